// RBFKANHead_71244917506725
// MI455X (gfx1250) — compile-verified
//
#include <hip/hip_runtime.h>
#include <math.h>

typedef __attribute__((ext_vector_type(2))) float v2f;
typedef __attribute__((ext_vector_type(8))) float v8f;

#define HW      65536        // H*W = 256*256
#define NTILES  32768        // B*H*W / 16 pixels per tile

// Fused RBF-KAN head:
//   z   = w_in(8x64) @ x_tile(64x16)          -> WMMA f32 16x16x4, 16 k-steps
//   phi = exp(-((z - c)/w)^2)                 -> VALU (32 expf / lane)
//   out = out_w(16x64) @ phi(64x16) + bias    -> WMMA f32 16x16x4, 16 k-steps
__global__ __launch_bounds__(256) void rbfkan_fused(
    const float* __restrict__ x,          // (8, 64, 256, 256)
    const float* __restrict__ w_in,       // (8, 64)
    const float* __restrict__ b_in,       // (8,)
    const float* __restrict__ centers,    // (1, 8, 8, 1, 1) -> flat 64
    const float* __restrict__ log_widths, // (1, 8, 8, 1, 1) -> flat 64
    const float* __restrict__ out_w,      // (16, 8, 8)      -> flat 16x64
    const float* __restrict__ out_bias,   // (16,)
    float* __restrict__ out)              // (8, 16, 256, 256)
{
    const int lane = (int)(threadIdx.x & 31);
    const int row  = lane & 15;   // matrix row (A) / pixel (B,C,D)
    const int half = lane >> 4;   // which K pair this lane holds
    const int wave   = (int)(blockIdx.x * (blockDim.x >> 5) + (threadIdx.x >> 5));
    const int nwaves = (int)(gridDim.x * (blockDim.x >> 5));

    // ---- loop-invariant fragments, preloaded once ----
    v2f a1[16];    // w_in fragments (rows 8..15 zero)
    v2f a2[16];    // out_weight fragments
    v2f cen[16];   // RBF centers for this lane's phi rows
    v2f invw[16];  // 1 / (softplus(log_width) + 1e-4)
    float bin[8], obias[8];

#pragma unroll
    for (int kc = 0; kc < 16; ++kc) {
        const int c0 = 4 * kc + 2 * half;          // K index pair {c0, c0+1}
        // w_in padded to 16 rows: clamp address in-bounds, select-to-zero.
        const int rr = row & 7;
        const float w0v = w_in[rr * 64 + c0];
        const float w1v = w_in[rr * 64 + c0 + 1];
        a1[kc].x = (row < 8) ? w0v : 0.0f;
        a1[kc].y = (row < 8) ? w1v : 0.0f;
        a2[kc].x = out_w[row * 64 + c0];
        a2[kc].y = out_w[row * 64 + c0 + 1];
        cen[kc].x = centers[c0];
        cen[kc].y = centers[c0 + 1];
        const float sp0 = log1pf(__expf(log_widths[c0]))     + 1.0e-4f;
        const float sp1 = log1pf(__expf(log_widths[c0 + 1])) + 1.0e-4f;
        invw[kc].x = 1.0f / sp0;
        invw[kc].y = 1.0f / sp1;
    }
#pragma unroll
    for (int m = 0; m < 8; ++m) bin[m] = b_in[m];
#pragma unroll
    for (int r = 0; r < 8; ++r) obias[r] = out_bias[r + 8 * half];

    // ---- grid-stride over 16-pixel tiles ----
    for (int t = wave; t < NTILES; t += nwaves) {
        const int g = t << 4;                 // global pixel index
        const int b = g >> 16;                // batch (HW = 65536 pixels)
        const int p = g & (HW - 1);           // pixel within image
        const float* xb = x + (size_t)b * 64u * HW + (size_t)p + (size_t)row;

        // Stage 1: z = w_in @ x_tile   (accumulate 16 WMMA, K=4 each)
        v8f z = {0.f, 0.f, 0.f, 0.f, 0.f, 0.f, 0.f, 0.f};
#pragma unroll
        for (int kc = 0; kc < 16; ++kc) {
            const int c0 = 4 * kc + 2 * half;
            v2f bf;
            bf.x = xb[(size_t)c0 * HW];
            bf.y = xb[(size_t)(c0 + 1) * HW];
            z = __builtin_amdgcn_wmma_f32_16x16x4_f32(
                    false, a1[kc], false, bf, (short)0, z, false, false);
        }

        // Broadcast the 8 valid z rows (held by lanes 0..15) to both halves,
        // then add the input bias.
        float zv[8];
#pragma unroll
        for (int m = 0; m < 8; ++m) {
            const int zi = __builtin_amdgcn_ds_bpermute(
                row << 2, __float_as_int(z[m]));
            zv[m] = __int_as_float(zi) + bin[m];
        }

        // Stage 2+3: phi = exp(-((z-c)/w)^2) built per k-step, fed to WMMA.
        // phi row = 4*kc + 2*half + j; its hid index (row>>3) == kc>>1 for
        // every half/j, so register arrays stay statically indexed.
        v8f o = {0.f, 0.f, 0.f, 0.f, 0.f, 0.f, 0.f, 0.f};
#pragma unroll
        for (int kc = 0; kc < 16; ++kc) {
            const float zm = zv[kc >> 1];
            const float t0 = (zm - cen[kc].x) * invw[kc].x;
            const float t1 = (zm - cen[kc].y) * invw[kc].y;
            v2f ph;
            ph.x = __expf(-t0 * t0);
            ph.y = __expf(-t1 * t1);
            o = __builtin_amdgcn_wmma_f32_16x16x4_f32(
                    false, a2[kc], false, ph, (short)0, o, false, false);
        }

        // Store: VGPR r, lanes 0-15 -> out row r; lanes 16-31 -> row r+8.
        float* ob = out + (size_t)b * 16u * HW + (size_t)p + (size_t)row;
#pragma unroll
        for (int r = 0; r < 8; ++r) {
            ob[(size_t)(r + 8 * half) * HW] = o[r] + obias[r];
        }
    }
}

extern "C" void kernel_launch(void* const* d_in, const int* in_sizes, int n_in,
                              void* d_out, int out_size, void* d_ws, size_t ws_size,
                              hipStream_t stream) {
    (void)in_sizes; (void)n_in; (void)d_ws; (void)ws_size; (void)out_size;
    const float* x          = (const float*)d_in[0];
    const float* w_in       = (const float*)d_in[1];
    const float* b_in       = (const float*)d_in[2];
    const float* centers    = (const float*)d_in[3];
    const float* log_widths = (const float*)d_in[4];
    const float* out_weight = (const float*)d_in[5];
    const float* out_bias   = (const float*)d_in[6];
    float* out = (float*)d_out;

    dim3 grid(1024);   // 8192 waves -> 4 tiles per wave
    dim3 block(256);   // 8 wave32 per workgroup
    hipLaunchKernelGGL(rbfkan_fused, grid, block, 0, stream,
                       x, w_in, b_in, centers, log_widths, out_weight,
                       out_bias, out);
}